// GCNLayer_15685220565555
// MI455X (gfx1250) — compile-verified
//
#include <hip/hip_runtime.h>
#include <math.h>

#define D 128
#define BN_EPS 1e-5f
#define XS_STRIDE 132   // 132 % 64 == 4 -> conflict-free A-operand LDS reads

typedef __attribute__((ext_vector_type(2))) float v2f;
typedef __attribute__((ext_vector_type(8))) float v8f;

// ---------------------------------------------------------------- init
// deg[i] = 1 (self loop), zero the BN partial-sum buffers.
__global__ void k_init(float* __restrict__ deg, float* __restrict__ sum,
                       float* __restrict__ sumsq, int n) {
    int i = blockIdx.x * blockDim.x + threadIdx.x;
    if (i < n) deg[i] = 1.0f;
    if (i < D) { sum[i] = 0.0f; sumsq[i] = 0.0f; }
}

// ---------------------------------------------------------------- degree
// deg[row[e]] += 1  (reference uses ones_like(ew), not ew)
__global__ void k_degree(const int* __restrict__ row, float* __restrict__ deg, int e) {
    int i = blockIdx.x * blockDim.x + threadIdx.x;
    if (i < e) atomicAdd(&deg[row[i]], 1.0f);
}

// ---------------------------------------------------------------- dinv
__global__ void k_dinv(float* __restrict__ deg, int n) {
    int i = blockIdx.x * blockDim.x + threadIdx.x;
    if (i < n) deg[i] = rsqrtf(deg[i]);   // deg >= 1 always (self loop)
}

// ---------------------------------------------------------------- GEMM: h = x @ W + b
// One block = 16 rows of x; 8 waves, wave w -> 16x16 tile at cols 16w.
// f32 WMMA 16x16x4, K-loop over 128 in steps of 4.
// Epilogue fuses the self-loop term: out = dinv^2 * h  (first write to out).
__global__ __launch_bounds__(256) void k_gemm(const float* __restrict__ x,
                                              const float* __restrict__ W,
                                              const float* __restrict__ bias,
                                              const float* __restrict__ dinv,
                                              float* __restrict__ h,
                                              float* __restrict__ out, int n) {
    __shared__ float xs[16 * XS_STRIDE];
    __shared__ float dsq[16];                 // dinv^2 per tile row
    const int tid = threadIdx.x;
    const int rowBase = blockIdx.x * 16;

    // cooperative load of the 16x128 x tile (zero-pad past n)
    for (int t = tid; t < 16 * D; t += 256) {
        int r = t >> 7, c = t & (D - 1);
        int gr = rowBase + r;
        xs[r * XS_STRIDE + c] = (gr < n) ? x[(size_t)gr * D + c] : 0.0f;
    }
    if (tid < 16) {
        int gr = rowBase + tid;
        float s = (gr < n) ? dinv[gr] : 0.0f;
        dsq[tid] = s * s;
    }
    __syncthreads();

    const int wave = tid >> 5;          // 0..7 -> column tile
    const int lane = tid & 31;
    const int colBase = wave * 16;
    const int nl = lane & 15;           // N (and M for A-operand) within tile
    const int koff = (lane < 16) ? 0 : 2;

    v8f c = {};
    #pragma unroll 4
    for (int k = 0; k < D; k += 4) {
        v2f a, b;
        // A 16x4 (MxK): lane m holds (K=koff, K=koff+1)
        a.x = xs[nl * XS_STRIDE + k + koff];
        a.y = xs[nl * XS_STRIDE + k + koff + 1];
        // B 4x16 (KxN): lane n holds (K=koff, K=koff+1)
        b.x = W[(size_t)(k + koff) * D + colBase + nl];
        b.y = W[(size_t)(k + koff + 1) * D + colBase + nl];
        c = __builtin_amdgcn_wmma_f32_16x16x4_f32(
                /*neg_a=*/false, a, /*neg_b=*/false, b,
                /*c_mod=*/(short)0, c, /*reuse_a=*/false, /*reuse_b=*/false);
    }

    const float bv = bias[colBase + nl];
    const int mBase = (lane < 16) ? 0 : 8;   // C/D layout: VGPR r -> M=r (lanes 0-15), M=r+8 (16-31)
    const int base = (rowBase + mBase) * D + colBase + nl;  // flat index of element r=0
    float* __restrict__ hp = h + base;
    float* __restrict__ op = out + base;

    if (rowBase + 16 <= n) {             // full tile: straight-line stores
        #pragma unroll
        for (int r = 0; r < 8; ++r) {
            float v = c[r] + bv;
            hp[r * D] = v;
            op[r * D] = dsq[mBase + r] * v;
        }
    } else {                             // ragged tail tile
        #pragma unroll
        for (int r = 0; r < 8; ++r) {
            if (rowBase + mBase + r < n) {
                float v = c[r] + bv;
                hp[r * D] = v;
                op[r * D] = dsq[mBase + r] * v;
            }
        }
    }
}

// ---------------------------------------------------------------- edge scatter
// One wave per edge: lane holds 4 features. float4 gather + 4 f32 atomic adds.
__global__ __launch_bounds__(256) void k_scatter(const int* __restrict__ row,
                                                 const int* __restrict__ col,
                                                 const float* __restrict__ ew,
                                                 const float* __restrict__ dinv,
                                                 const float* __restrict__ h,
                                                 float* __restrict__ out, int e) {
    const int lane = threadIdx.x & 31;
    const int edge = blockIdx.x * (blockDim.x >> 5) + (threadIdx.x >> 5);
    if (edge >= e) return;
    const int r = row[edge];
    const int cdst = col[edge];
    const float norm = dinv[r] * dinv[cdst] * ew[edge];

    const float4 v = ((const float4*)(h + (size_t)r * D))[lane];
    float* o = out + (size_t)cdst * D + lane * 4;
    atomicAdd(o + 0, norm * v.x);
    atomicAdd(o + 1, norm * v.y);
    atomicAdd(o + 2, norm * v.z);
    atomicAdd(o + 3, norm * v.w);
}

// ---------------------------------------------------------------- BN column stats
// Block covers 256 rows; thread f accumulates feature f, then atomics to sum/sumsq.
__global__ __launch_bounds__(128) void k_stats(const float* __restrict__ out,
                                               float* __restrict__ sum,
                                               float* __restrict__ sumsq, int n) {
    const int f = threadIdx.x;
    const int r0 = blockIdx.x * 256;
    const int r1 = min(n, r0 + 256);
    float s = 0.0f, s2 = 0.0f;
    for (int r = r0; r < r1; ++r) {
        float v = out[(size_t)r * D + f];
        s += v;
        s2 += v * v;
    }
    atomicAdd(&sum[f], s);
    atomicAdd(&sumsq[f], s2);
}

// ---------------------------------------------------------------- BN scale/shift
__global__ void k_bnparams(const float* __restrict__ sum, const float* __restrict__ sumsq,
                           const float* __restrict__ gamma, const float* __restrict__ beta,
                           float* __restrict__ scale, float* __restrict__ shift, float invN) {
    int f = threadIdx.x;
    if (f < D) {
        float mean = sum[f] * invN;
        float var = sumsq[f] * invN - mean * mean;   // biased batch variance
        float s = gamma[f] * rsqrtf(var + BN_EPS);
        scale[f] = s;
        shift[f] = beta[f] - mean * s;
    }
}

// ---------------------------------------------------------------- BN apply + ReLU (in place)
__global__ void k_bnrelu(float* __restrict__ out, const float* __restrict__ scale,
                         const float* __restrict__ shift, int total) {
    int idx = blockIdx.x * blockDim.x + threadIdx.x;
    if (idx < total) {
        int f = idx & (D - 1);
        float v = out[idx] * scale[f] + shift[f];
        out[idx] = v > 0.0f ? v : 0.0f;
    }
}

extern "C" void kernel_launch(void* const* d_in, const int* in_sizes, int n_in,
                              void* d_out, int out_size, void* d_ws, size_t ws_size,
                              hipStream_t stream) {
    const float* x     = (const float*)d_in[0];
    const int*   eidx  = (const int*)  d_in[1];
    const float* ew    = (const float*)d_in[2];
    const float* W     = (const float*)d_in[3];
    const float* bias  = (const float*)d_in[4];
    const float* gamma = (const float*)d_in[5];
    const float* beta  = (const float*)d_in[6];
    float* out = (float*)d_out;

    const int n = in_sizes[0] / D;          // 100000
    const int e = in_sizes[2];              // 1600000
    const int* row = eidx;                  // edge_index[0] = source
    const int* col = eidx + e;              // edge_index[1] = destination

    // workspace layout (floats)
    float* h     = (float*)d_ws;            // n*D
    float* deg   = h + (size_t)n * D;       // n   (becomes dinv in place)
    float* sum   = deg + n;                 // D
    float* sumsq = sum + D;                 // D
    float* scale = sumsq + D;               // D
    float* shift = scale + D;               // D

    const int total = n * D;
    const int tpb = 256;

    k_init    <<<(n + tpb - 1) / tpb, tpb, 0, stream>>>(deg, sum, sumsq, n);
    k_degree  <<<(e + tpb - 1) / tpb, tpb, 0, stream>>>(row, deg, e);
    k_dinv    <<<(n + tpb - 1) / tpb, tpb, 0, stream>>>(deg, n);
    k_gemm    <<<(n + 15) / 16, 256, 0, stream>>>(x, W, bias, deg, h, out, n);
    k_scatter <<<(e + 7) / 8, 256, 0, stream>>>(row, col, ew, deg, h, out, e);
    k_stats   <<<(n + 255) / 256, 128, 0, stream>>>(out, sum, sumsq, n);
    k_bnparams<<<1, 128, 0, stream>>>(sum, sumsq, gamma, beta, scale, shift, 1.0f / (float)n);
    k_bnrelu  <<<(total + tpb - 1) / tpb, tpb, 0, stream>>>(out, scale, shift, total);
}